// PINN_66236985638930
// MI455X (gfx1250) — compile-verified
//
#include <hip/hip_runtime.h>
#include <hip/hip_bf16.h>
#include <math.h>

typedef __attribute__((ext_vector_type(2))) float v2f;
typedef __attribute__((ext_vector_type(8))) float v8f;

#define HID   128
#define LDH   130            // LDS row stride (pad 2 -> conflict-free A loads)
#define ROWS  16             // points per block
#define NPTS  32768
#define NIC   16384
#define IGM1  2.5f           // 1/(gamma-1), gamma=1.4

// ---------------- WMMA helpers (V_WMMA_F32_16X16X4_F32, f32 in / f32 acc) ---
// A 16x4 layout (ISA 7.12.2): lane l holds A[m=l&15][k0], A[m][k0+1], k0 = 2*(l>>4).
// B 4x16 (analogous split): lane l holds B[k0][n=l&15], B[k0+1][n].
// C/D 16x16: vgpr j, lane l -> C[j + 8*(l>>4)][l&15].

__device__ __forceinline__ void load_bfrag(const float* __restrict__ W, int ncol,
                                           int lane, v2f* __restrict__ b) {
  const int n  = ncol + (lane & 15);
  const int kh = (lane >> 4) * 2;
#pragma unroll
  for (int s = 0; s < 32; ++s) {
    const int k = 4 * s + kh;
    v2f t;
    t.x = W[k * HID + n];
    t.y = W[(k + 1) * HID + n];
    b[s] = t;
  }
}

__device__ __forceinline__ v8f wmma_gemm(const float* __restrict__ buf,
                                         const v2f* __restrict__ b, int lane) {
  v8f c = {};
  const int m  = lane & 15;
  const int ko = (lane >> 4) * 2;
#pragma unroll
  for (int s = 0; s < 32; ++s) {
    v2f a = *(const v2f*)(buf + m * LDH + 4 * s + ko);   // 8B-aligned ds_load_b64
    c = __builtin_amdgcn_wmma_f32_16x16x4_f32(false, a, false, b[s],
                                              (short)0, c, false, false);
  }
  return c;
}

// ---------------- fused residual kernel (m-net 6 streams + n-net) -----------
__global__ __launch_bounds__(256)
void pinn_residual(const float* __restrict__ xtr,
                   const float* __restrict__ mw0, const float* __restrict__ mb0,
                   const float* __restrict__ mw1, const float* __restrict__ mb1,
                   const float* __restrict__ mw2, const float* __restrict__ mb2,
                   const float* __restrict__ mw3, const float* __restrict__ mb3,
                   const float* __restrict__ nw0, const float* __restrict__ nb0,
                   const float* __restrict__ nw1, const float* __restrict__ nb1,
                   const float* __restrict__ nw2, const float* __restrict__ nb2,
                   const float* __restrict__ nw3, const float* __restrict__ nb3,
                   float* __restrict__ ws) {
  __shared__ float hv [ROWS * LDH];
  __shared__ float hx [ROWS * LDH];
  __shared__ float hy [ROWS * LDH];
  __shared__ float ht [ROWS * LDH];
  __shared__ float hxx[ROWS * LDH];
  __shared__ float hyy[ROWS * LDH];
  __shared__ float hn [ROWS * LDH];
  __shared__ float px[ROWS], py[ROWS], ptm[ROWS];
  __shared__ float outm[6 * 64];      // [stream][row][4]
  __shared__ float outn[ROWS];
  __shared__ float red[ROWS * 5];

  const int tid  = threadIdx.x;
  const int lane = tid & 31;
  const int wv   = tid >> 5;
  const int ncol = wv * 16;
  const int base = blockIdx.x * ROWS;

  if (tid < ROWS) {
    px [tid] = xtr[(base + tid) * 3 + 0];
    py [tid] = xtr[(base + tid) * 3 + 1];
    ptm[tid] = xtr[(base + tid) * 3 + 2];
  }
  __syncthreads();

  // ---- layer 0 (3 -> 128), elementwise second-order forward seed -----------
#pragma unroll
  for (int i = 0; i < 8; ++i) {
    const int idx = tid + i * 256;
    const int r = idx >> 7, c = idx & 127;
    const float w0 = mw0[c], w1 = mw0[HID + c], w2 = mw0[2 * HID + c];
    const float z  = px[r] * w0 + py[r] * w1 + ptm[r] * w2 + mb0[c];
    const float a  = tanhf(z);
    const float g1 = 1.f - a * a;
    const float g2 = -2.f * a * g1;
    hv [r * LDH + c] = a;
    hx [r * LDH + c] = g1 * w0;
    hy [r * LDH + c] = g1 * w1;
    ht [r * LDH + c] = g1 * w2;
    hxx[r * LDH + c] = g2 * w0 * w0;
    hyy[r * LDH + c] = g2 * w1 * w1;
    const float zn = px[r] * nw0[c] + py[r] * nw0[HID + c] +
                     ptm[r] * nw0[2 * HID + c] + nb0[c];
    hn[r * LDH + c] = tanhf(zn);
  }
  __syncthreads();

  // ---- hidden layers 1..2 (128 -> 128): 7 WMMA GEMM streams ---------------
  const float* MW[2] = {mw1, mw2};
  const float* MB[2] = {mb1, mb2};
  const float* NW[2] = {nw1, nw2};
  const float* NB[2] = {nb1, nb2};
  for (int L = 0; L < 2; ++L) {
    v2f bf[32];
    // n-net value stream first so bf can be reused for the m-net weights
    load_bfrag(NW[L], ncol, lane, bf);
    v8f zn = wmma_gemm(hn, bf, lane);
    load_bfrag(MW[L], ncol, lane, bf);
    v8f zv = wmma_gemm(hv, bf, lane);
    __syncthreads();                       // all phase-A reads of hv/hn done

    const int cn = ncol + (lane & 15);
    const int mh = (lane >> 4) * 8;
    const float bb = MB[L][cn], nbb = NB[L][cn];
    float av[8];
#pragma unroll
    for (int j = 0; j < 8; ++j) {
      const float a = tanhf(zv[j] + bb);
      av[j] = a;
      hv[(mh + j) * LDH + cn] = a;
      hn[(mh + j) * LDH + cn] = tanhf(zn[j] + nbb);
    }

    // phase B: five tangent streams, same B fragments
    v8f zx  = wmma_gemm(hx,  bf, lane);
    v8f zxxv= wmma_gemm(hxx, bf, lane);
    v8f zy  = wmma_gemm(hy,  bf, lane);
    v8f zyyv= wmma_gemm(hyy, bf, lane);
    v8f zt  = wmma_gemm(ht,  bf, lane);
    __syncthreads();                       // all reads of old tangents done

#pragma unroll
    for (int j = 0; j < 8; ++j) {
      const float a  = av[j];
      const float g1 = 1.f - a * a;
      const float g2 = -2.f * a * g1;
      const int   o  = (mh + j) * LDH + cn;
      const float sx = zx[j], sy = zy[j];
      hx [o] = g1 * sx;
      hxx[o] = g2 * sx * sx + g1 * zxxv[j];
      hy [o] = g1 * sy;
      hyy[o] = g2 * sy * sy + g1 * zyyv[j];
      ht [o] = g1 * zt[j];
    }
    __syncthreads();
  }

  // ---- output layers (128->4 for 6 streams, 128->1 for n) ------------------
  {
    const float* bufs[6] = {hv, hx, hy, ht, hxx, hyy};
    for (int v = tid; v < 384; v += 256) {
      const int s = v >> 6, rem = v & 63, r = rem >> 2, c = rem & 3;
      const float* bp = bufs[s];
      float acc = (s == 0) ? mb3[c] : 0.f;
      for (int k = 0; k < HID; ++k) acc += bp[r * LDH + k] * mw3[k * 4 + c];
      outm[v] = acc;
    }
    if (tid < ROWS) {
      float acc = nb3[0];
      for (int k = 0; k < HID; ++k) acc += hn[tid * LDH + k] * nw3[k];
      outn[tid] = acc;
    }
  }
  __syncthreads();

  // ---- Euler residual chain rule per point ---------------------------------
  if (tid < ROWS) {
    const int r = tid;
#define OM(s, c) outm[(s) * 64 + r * 4 + (c)]
    const float rho = OM(0,0), pr  = OM(0,1), uu  = OM(0,2), vv  = OM(0,3);
    const float rx  = OM(1,0), pxd = OM(1,1), ux  = OM(1,2), vx  = OM(1,3);
    const float ry  = OM(2,0), pyd = OM(2,1), uy  = OM(2,2), vy  = OM(2,3);
    const float rt  = OM(3,0), ptd = OM(3,1), ut  = OM(3,2), vt  = OM(3,3);
    const float rxx = OM(4,0), pxx = OM(4,1), uxx = OM(4,2), vxx = OM(4,3);
    const float ryy = OM(5,0), pyy = OM(5,1), uyy = OM(5,2), vyy = OM(5,3);
#undef OM
    const float ke = 0.5f * (uu * uu + vv * vv);
    const float E  = pr * IGM1 + rho * ke;
    const float Ex = pxd * IGM1 + rx * ke + rho * (uu * ux + vv * vx);
    const float Ey = pyd * IGM1 + ry * ke + rho * (uu * uy + vv * vy);
    const float Et = ptd * IGM1 + rt * ke + rho * (uu * ut + vv * vt);
    const float Ep = E + pr;
    // U_t
    const float Ut0 = rt;
    const float Ut1 = rt * uu + rho * ut;
    const float Ut2 = rt * vv + rho * vt;
    const float Ut3 = Et;
    // f_x
    const float Fx0 = rx * uu + rho * ux;
    const float Fx1 = rx * uu * uu + 2.f * rho * uu * ux + pxd;
    const float Fx2 = rx * uu * vv + rho * ux * vv + rho * uu * vx;
    const float Fx3 = ux * Ep + uu * (Ex + pxd);
    // g_y
    const float Gy0 = ry * vv + rho * vy;
    const float Gy1 = ry * uu * vv + rho * uy * vv + rho * uu * vy;
    const float Gy2 = ry * vv * vv + 2.f * rho * vv * vy + pyd;
    const float Gy3 = vy * Ep + vv * (Ey + pyd);
    // Laplacian of U
    const float Exx = pxx * IGM1 + rxx * ke + 2.f * rx * (uu * ux + vv * vx) +
                      rho * (ux * ux + uu * uxx + vx * vx + vv * vxx);
    const float Eyy = pyy * IGM1 + ryy * ke + 2.f * ry * (uu * uy + vv * vy) +
                      rho * (uy * uy + uu * uyy + vy * vy + vv * vyy);
    const float L0 = rxx + ryy;
    const float L1 = (rxx * uu + 2.f * rx * ux + rho * uxx) +
                     (ryy * uu + 2.f * ry * uy + rho * uyy);
    const float L2 = (rxx * vv + 2.f * rx * vx + rho * vxx) +
                     (ryy * vv + 2.f * ry * vy + rho * vyy);
    const float L3 = Exx + Eyy;
    const float nno = outn[r];
    const float mu  = 0.01f * nno * nno;
    const float r0 = Ut0 + Fx0 + Gy0 - mu * L0;
    const float r1 = Ut1 + Fx1 + Gy1 - mu * L1;
    const float r2 = Ut2 + Fx2 + Gy2 - mu * L2;
    const float r3 = Ut3 + Fx3 + Gy3 - mu * L3;
    red[r * 5 + 0] = r0 * r0;
    red[r * 5 + 1] = r1 * r1;
    red[r * 5 + 2] = r2 * r2;
    red[r * 5 + 3] = r3 * r3;
    red[r * 5 + 4] = mu * mu;
  }
  __syncthreads();
  if (tid == 0) {
    float s0 = 0, s1 = 0, s2 = 0, s3 = 0, s4 = 0;
    for (int r = 0; r < ROWS; ++r) {
      s0 += red[r * 5 + 0]; s1 += red[r * 5 + 1]; s2 += red[r * 5 + 2];
      s3 += red[r * 5 + 3]; s4 += red[r * 5 + 4];
    }
    atomicAdd(ws + 0, s0); atomicAdd(ws + 1, s1); atomicAdd(ws + 2, s2);
    atomicAdd(ws + 3, s3); atomicAdd(ws + 4, s4);
  }
}

// ---------------- IC quadrant loss kernel (value stream only) ---------------
__global__ __launch_bounds__(256)
void pinn_ic(const float* __restrict__ xin, const float* __restrict__ ic,
             const float* __restrict__ mw0, const float* __restrict__ mb0,
             const float* __restrict__ mw1, const float* __restrict__ mb1,
             const float* __restrict__ mw2, const float* __restrict__ mb2,
             const float* __restrict__ mw3, const float* __restrict__ mb3,
             float* __restrict__ ws) {
  __shared__ float hv[ROWS * LDH];
  __shared__ float pxs[ROWS], pys[ROWS], pts[ROWS];
  __shared__ float outm[64];
  __shared__ float redq[ROWS];
  __shared__ int   qrow[ROWS];

  const int tid  = threadIdx.x;
  const int lane = tid & 31;
  const int wv   = tid >> 5;
  const int ncol = wv * 16;
  const int base = blockIdx.x * ROWS;

  if (tid < ROWS) {
    pxs[tid] = xin[(base + tid) * 3 + 0];
    pys[tid] = xin[(base + tid) * 3 + 1];
    pts[tid] = xin[(base + tid) * 3 + 2];
  }
  __syncthreads();

#pragma unroll
  for (int i = 0; i < 8; ++i) {
    const int idx = tid + i * 256;
    const int r = idx >> 7, c = idx & 127;
    const float z = pxs[r] * mw0[c] + pys[r] * mw0[HID + c] +
                    pts[r] * mw0[2 * HID + c] + mb0[c];
    hv[r * LDH + c] = tanhf(z);
  }
  __syncthreads();

  const float* MW[2] = {mw1, mw2};
  const float* MB[2] = {mb1, mb2};
  for (int L = 0; L < 2; ++L) {
    v2f bf[32];
    load_bfrag(MW[L], ncol, lane, bf);
    v8f zv = wmma_gemm(hv, bf, lane);
    __syncthreads();
    const int cn = ncol + (lane & 15);
    const int mh = (lane >> 4) * 8;
    const float bb = MB[L][cn];
#pragma unroll
    for (int j = 0; j < 8; ++j) hv[(mh + j) * LDH + cn] = tanhf(zv[j] + bb);
    __syncthreads();
  }

  if (tid < 64) {
    const int r = tid >> 2, c = tid & 3;
    float acc = mb3[c];
    for (int k = 0; k < HID; ++k) acc += hv[r * LDH + k] * mw3[k * 4 + c];
    outm[tid] = acc;
  }
  __syncthreads();

  if (tid < ROWS) {
    const float xv = pxs[tid], yv = pys[tid];
    const bool xl = xv < 0.5f, xr = xv > 0.5f;
    const bool ytp = yv > 0.5f, yb = yv < 0.5f;
    int row = -1;
    if (xl && ytp) row = 1;
    else if (xr && ytp) row = 3;
    else if (xl && yb) row = 0;
    else if (xr && yb) row = 2;
    float sq = 0.f;
    if (row >= 0) {
      // perm = [0,3,1,2] applied to ic[row]
      const float t0 = ic[row * 4 + 0], t1 = ic[row * 4 + 3];
      const float t2 = ic[row * 4 + 1], t3 = ic[row * 4 + 2];
      const float d0 = outm[tid * 4 + 0] - t0;
      const float d1 = outm[tid * 4 + 1] - t1;
      const float d2 = outm[tid * 4 + 2] - t2;
      const float d3 = outm[tid * 4 + 3] - t3;
      sq = d0 * d0 + d1 * d1 + d2 * d2 + d3 * d3;
    }
    redq[tid] = sq;
    qrow[tid] = row;
  }
  __syncthreads();
  if (tid == 0) {
    float qs[4] = {0, 0, 0, 0};
    float qc[4] = {0, 0, 0, 0};
    for (int r = 0; r < ROWS; ++r) {
      const int q = qrow[r];
      if (q >= 0) { qs[q] += redq[r]; qc[q] += 1.f; }
    }
    for (int q = 0; q < 4; ++q) {
      if (qs[q] != 0.f || qc[q] != 0.f) {
        atomicAdd(ws + 5 + q, qs[q]);
        atomicAdd(ws + 9 + q, qc[q]);
      }
    }
  }
}

// ---------------- init / finalize ------------------------------------------
__global__ void pinn_init(float* ws) {
  if (threadIdx.x < 16) ws[threadIdx.x] = 0.f;
}

__global__ void pinn_finalize(const float* __restrict__ ws, float* __restrict__ out) {
  if (threadIdx.x == 0 && blockIdx.x == 0) {
    const float invN = 1.0f / (float)NPTS;
    const float res = (ws[0] + ws[1] + ws[2] + ws[3]) * invN;
    float init = 0.f;
    for (int q = 0; q < 4; ++q) {
      float d = ws[9 + q];
      if (d < 1.f) d = 1.f;
      init += ws[5 + q] / d;
    }
    out[0] = res + 10.f * init + 0.1f * ws[4] * invN;
  }
}

extern "C" void kernel_launch(void* const* d_in, const int* in_sizes, int n_in,
                              void* d_out, int out_size, void* d_ws, size_t ws_size,
                              hipStream_t stream) {
  const float* xtr = (const float*)d_in[0];
  const float* xin = (const float*)d_in[1];
  const float* ic  = (const float*)d_in[2];
  const float* mw0 = (const float*)d_in[3];  const float* mb0 = (const float*)d_in[4];
  const float* mw1 = (const float*)d_in[5];  const float* mb1 = (const float*)d_in[6];
  const float* mw2 = (const float*)d_in[7];  const float* mb2 = (const float*)d_in[8];
  const float* mw3 = (const float*)d_in[9];  const float* mb3 = (const float*)d_in[10];
  const float* nw0 = (const float*)d_in[11]; const float* nb0 = (const float*)d_in[12];
  const float* nw1 = (const float*)d_in[13]; const float* nb1 = (const float*)d_in[14];
  const float* nw2 = (const float*)d_in[15]; const float* nb2 = (const float*)d_in[16];
  const float* nw3 = (const float*)d_in[17]; const float* nb3 = (const float*)d_in[18];
  float* ws  = (float*)d_ws;
  float* out = (float*)d_out;

  pinn_init<<<1, 32, 0, stream>>>(ws);
  pinn_residual<<<NPTS / ROWS, 256, 0, stream>>>(
      xtr, mw0, mb0, mw1, mb1, mw2, mb2, mw3, mb3,
      nw0, nb0, nw1, nb1, nw2, nb2, nw3, nb3, ws);
  pinn_ic<<<NIC / ROWS, 256, 0, stream>>>(
      xin, ic, mw0, mb0, mw1, mb1, mw2, mb2, mw3, mb3, ws);
  pinn_finalize<<<1, 32, 0, stream>>>(ws, out);
}